// PatchBlender_3229815406881
// MI455X (gfx1250) — compile-verified
//
#include <hip/hip_runtime.h>
#include <hip/hip_bf16.h>

typedef __attribute__((ext_vector_type(2))) float v2f;
typedef __attribute__((ext_vector_type(8))) float v8f;

// Problem geometry (compile-time constants from the reference)
constexpr int BATCH   = 8;
constexpr int FRAMES  = 16;
constexpr int EMBED   = 768;
constexpr int PATCHES = 196;
constexpr int PE      = PATCHES * EMBED;        // 150528, contiguous "N" dim
constexpr int SEQ_E   = (FRAMES * PATCHES + 1) * EMBED; // 3137*768 per batch
constexpr int TILES_PER_B   = PE / 16;          // 9408 16-wide N tiles
constexpr int TILES_PER_WAVE = 8;               // amortize A-load, pipeline depth
constexpr int WAVES_PER_B   = TILES_PER_B / TILES_PER_WAVE; // 1176
constexpr int TOTAL_WAVES   = BATCH * WAVES_PER_B;          // 9408
constexpr int BLOCK_THREADS = 256;              // 8 wave32 per block

// out[b, t, n] = sum_f M[t,f] * x[b, f, n]   (n contiguous over 150528)
// One wave computes a 16(t) x 16(n) tile per inner iteration via
// V_WMMA_F32_16X16X4_F32, accumulating over 4 K-steps of 4 frames.
__global__ __launch_bounds__(BLOCK_THREADS)
void patch_blender_wmma(const float* __restrict__ x,
                        const float* __restrict__ M,
                        float* __restrict__ out)
{
    const int wave = (blockIdx.x * BLOCK_THREADS + threadIdx.x) >> 5;
    const int lane = threadIdx.x & 31;
    if (wave >= TOTAL_WAVES) return;

    const int b        = wave / WAVES_PER_B;
    const int wave_in_b = wave - b * WAVES_PER_B;
    const int n_base0  = wave_in_b * (16 * TILES_PER_WAVE);

    const int col = lane & 15;   // A: M-row (t) ; B/D: N column
    const int g   = lane >> 4;   // half-wave group (K / t split)

    // ---- A: smoothing matrix, loaded once, reused for all tiles ----
    // A 16x4 layout: lane = row t, VGPR0 holds K = {0 (lanes 0-15), 2 (16-31)},
    // VGPR1 holds K = {1, 3}; 4 K-steps cover K = 0..15.
    v2f a[4];
#pragma unroll
    for (int s = 0; s < 4; ++s) {
        const int k = s * 4 + 2 * g;
        a[s].x = M[col * 16 + k];
        a[s].y = M[col * 16 + k + 1];
    }

    const float* __restrict__ src = x   + (size_t)b * SEQ_E + EMBED; // skip CLS
    float*       __restrict__ dst = out + (size_t)b * SEQ_E + EMBED;

#pragma unroll
    for (int c = 0; c < TILES_PER_WAVE; ++c) {
        const int n = n_base0 + c * 16 + col;

        v8f acc = {};
#pragma unroll
        for (int s = 0; s < 4; ++s) {
            const int k = s * 4 + 2 * g;           // frame index for this half-wave
            // B 4x16 slab: VGPR0 row K=k, VGPR1 row K=k+1; lanes 0-15 and
            // 16-31 each read 16 contiguous floats (two 64B segments / load).
            v2f bb;
            bb.x = src[(size_t)k       * PE + n];
            bb.y = src[(size_t)(k + 1) * PE + n];
            acc = __builtin_amdgcn_wmma_f32_16x16x4_f32(
                /*neg_a=*/false, a[s],
                /*neg_b=*/false, bb,
                /*c_mod=*/(short)0, acc,
                /*reuse_a=*/false, /*reuse_b=*/false);
        }

        // D 16x16: VGPR r -> rows t=r (lanes 0-15) and t=r+8 (lanes 16-31)
#pragma unroll
        for (int r = 0; r < 8; ++r) {
            const int t = r + 8 * g;
            dst[(size_t)t * PE + n] = acc[r];
        }
    }
}

// Copy the CLS token: 8 * 768 floats.
__global__ void cls_copy_kernel(const float* __restrict__ x,
                                float* __restrict__ out)
{
    const int i = blockIdx.x * blockDim.x + threadIdx.x;
    if (i < BATCH * EMBED) {
        const int b = i / EMBED;
        const int e = i - b * EMBED;
        out[(size_t)b * SEQ_E + e] = x[(size_t)b * SEQ_E + e];
    }
}

extern "C" void kernel_launch(void* const* d_in, const int* in_sizes, int n_in,
                              void* d_out, int out_size, void* d_ws, size_t ws_size,
                              hipStream_t stream)
{
    const float* x = (const float*)d_in[0];            // (8, 3137, 768) f32
    const float* M = (const float*)d_in[1];            // (16, 16) f32
    float* out = (float*)d_out;                        // (8, 3137, 768) f32

    cls_copy_kernel<<<(BATCH * EMBED + 255) / 256, 256, 0, stream>>>(x, out);

    const int total_threads = TOTAL_WAVES * 32;        // 301056
    const int blocks = total_threads / BLOCK_THREADS;  // 1176
    patch_blender_wmma<<<blocks, BLOCK_THREADS, 0, stream>>>(x, M, out);
}